// DeformLoss_84756884619917
// MI455X (gfx1250) — compile-verified
//
#include <hip/hip_runtime.h>
#include <math.h>

typedef __attribute__((ext_vector_type(2))) float v2f;
typedef __attribute__((ext_vector_type(8))) float v8f;

struct F3 { float x, y, z; };

__device__ __forceinline__ F3 load3(const float* __restrict__ p, int i) {
    F3 r; const float* q = p + 3 * i;
    r.x = q[0]; r.y = q[1]; r.z = q[2];
    return r;
}
__device__ __forceinline__ F3 sub3(F3 a, F3 b) { F3 r{a.x-b.x, a.y-b.y, a.z-b.z}; return r; }
__device__ __forceinline__ float dot3(F3 a, F3 b) { return fmaf(a.x,b.x, fmaf(a.y,b.y, a.z*b.z)); }
__device__ __forceinline__ F3 cross3(F3 a, F3 b) {
    F3 r;
    r.x = fmaf(a.y, b.z, -a.z*b.y);
    r.y = fmaf(a.z, b.x, -a.x*b.z);
    r.z = fmaf(a.x, b.y, -a.y*b.x);
    return r;
}

// Reduce 256 f32 values in LDS to a single f32 using V_WMMA_F32_16X16X4_F32.
// Called by wave 0 only (all 32 lanes active -> EXEC all ones, as WMMA requires).
// A (16x4 f32, 2 VGPRs): VGPR0 holds K=0 (lanes 0-15) / K=2 (lanes 16-31),
// VGPR1 holds K=1 / K=3. B = all-ones 4x16, so D[m,n] = sum_k A[m,k], chained
// over 4 tiles via the C accumulator. Column n is identical for all n; rows
// 0-7 live in lanes 0-15 (VGPRs 0-7), rows 8-15 in lanes 16-31.
__device__ __forceinline__ float wmma_reduce256(const float* __restrict__ sdata, int lane) {
    v2f ones; ones.x = 1.0f; ones.y = 1.0f;
    v8f acc = {};
    const int base = ((lane >= 16) ? 32 : 0) + (lane & 15);
#pragma unroll
    for (int t = 0; t < 4; ++t) {
        v2f a;
        a.x = sdata[t * 64 + base];
        a.y = sdata[t * 64 + base + 16];
        acc = __builtin_amdgcn_wmma_f32_16x16x4_f32(
            /*neg_a=*/false, a, /*neg_b=*/false, ones,
            /*c_mod=*/(short)0, acc, /*reuse_a=*/false, /*reuse_b=*/false);
    }
    float s = acc[0] + acc[1] + acc[2] + acc[3] + acc[4] + acc[5] + acc[6] + acc[7];
    s += __shfl_xor(s, 16, 32);   // add rows 8-15 half to rows 0-7 half
    return s;                      // all lanes hold the full 256-value sum
}

__global__ __launch_bounds__(256)
void deform_partial(const float* __restrict__ vs, const float* __restrict__ vt,
                    const int* __restrict__ face, const float* __restrict__ rw,
                    float* __restrict__ block_sums, int nf) {
    // softmax(raw_weights): w0=shear, w1=scale, w2=bend (L2-cached 12B read)
    float r0 = rw[0], r1 = rw[1], r2 = rw[2];
    float mx = fmaxf(r0, fmaxf(r1, r2));
    float e0 = expf(r0 - mx), e1 = expf(r1 - mx), e2 = expf(r2 - mx);
    float inv = 1.0f / (e0 + e1 + e2);
    float k_shear = e0 * inv;
    float k_scale = 0.1f  * e1 * inv;
    float k_bend  = 0.01f * e2 * inv;

    const int stride = gridDim.x * blockDim.x;
    float accum = 0.0f;
    for (int n = blockIdx.x * blockDim.x + threadIdx.x; n < nf; n += stride) {
        int nn = n + stride;
        if (nn < nf) __builtin_prefetch(face + 3 * nn, 0, 3);  // global_prefetch_b8

        int i0 = face[3 * n + 0];
        int i1 = face[3 * n + 1];
        int i2 = face[3 * n + 2];

        F3 s0 = load3(vs, i0), s1 = load3(vs, i1), s2 = load3(vs, i2);
        F3 t0 = load3(vt, i0), t1 = load3(vt, i1), t2 = load3(vt, i2);

        F3 u = sub3(s1, s0);
        F3 v = sub3(s2, s0);
        F3 p = sub3(u, sub3(t1, t0));   // of_diff column 0
        F3 q = sub3(v, sub3(t2, t0));   // of_diff column 1

        float a = dot3(u, u), b = dot3(u, v), c = dot3(v, v);
        float det = fmaf(a, c, -b * b);
        F3  N  = cross3(u, v);
        float n2 = dot3(N, N);
        float area = 0.5f * sqrtf(n2);

        float np = dot3(p, N), nq = dot3(q, N);
        float pu = dot3(p, u), pv = dot3(p, v);
        float qu = dot3(q, u), qv = dot3(q, v);

        float D00 = 2.0f * pu;
        float D01 = pv + qu;
        float D11 = 2.0f * qv;

        float invdet = 1.0f / det;
        // (I-P) quadratic form: (c*np^2 - 2b*np*nq + a*nq^2) / det^2
        float norm_loss = (fmaf(c * np, np, fmaf(-2.0f * b * np, nq, a * nq * nq)))
                          * invdet * invdet;
        float stretch = fmaf(c, D00, fmaf(-2.0f * b, D01, a * D11)) * invdet;  // trace(inv_mt*distort)
        float shear   = fmaf(a + c, D01, -b * (D00 + D11)) * invdet;           // off-diag sum

        float loss = fmaf(k_bend, norm_loss, fmaf(k_scale, stretch, k_shear * shear));
        accum = fmaf(loss, area, accum);
    }

    __shared__ float sdata[256];
    sdata[threadIdx.x] = accum;
    __syncthreads();

    if (threadIdx.x < 32) {
        float tot = wmma_reduce256(sdata, (int)threadIdx.x);
        if (threadIdx.x == 0) block_sums[blockIdx.x] = tot;
    }
}

__global__ __launch_bounds__(256)
void deform_final(const float* __restrict__ block_sums, int nblocks,
                  float* __restrict__ out, float scale) {
    float accum = 0.0f;
    for (int i = threadIdx.x; i < nblocks; i += 256) accum += block_sums[i];

    __shared__ float sdata[256];
    sdata[threadIdx.x] = accum;
    __syncthreads();

    if (threadIdx.x < 32) {
        float tot = wmma_reduce256(sdata, (int)threadIdx.x);
        if (threadIdx.x == 0) out[0] = tot * scale;   // LOSS_WEIGHT == 1.0
    }
}

extern "C" void kernel_launch(void* const* d_in, const int* in_sizes, int n_in,
                              void* d_out, int out_size, void* d_ws, size_t ws_size,
                              hipStream_t stream) {
    const float* vs   = (const float*)d_in[0];   // vert_s  (NV,3) f32
    const float* vt   = (const float*)d_in[1];   // vert_t  (NV,3) f32
    const int*   face = (const int*)  d_in[2];   // face    (NF,3) i32
    const float* rw   = (const float*)d_in[3];   // raw_weights (3,) f32

    const int nf = in_sizes[2] / 3;
    const int NT = 256;
    const int NB = 1024;                          // 1024 block partials -> 4 KB of d_ws

    float* block_sums = (float*)d_ws;

    deform_partial<<<NB, NT, 0, stream>>>(vs, vt, face, rw, block_sums, nf);
    deform_final<<<1, NT, 0, stream>>>(block_sums, NB, (float*)d_out,
                                       1.0f / (float)nf);
}